// Deformable_Conv2d_56066503082058
// MI455X (gfx1250) — compile-verified
//
#include <hip/hip_runtime.h>
#include <hip/hip_bf16.h>
#include <math.h>

typedef float v2f __attribute__((ext_vector_type(2)));
typedef float v8f __attribute__((ext_vector_type(8)));

// Geometry constants for this problem instance
// x: (4, 256, 64, 64)  w_p: (18,256,3,3)  b_p:(18)  w_m:(9,256,3,3)
// w_ad:(3,256,3,3)     w_conv:(256,256,3,3)  out:(4,256,64,64)
#define B_   4
#define C_   256
#define H_   64
#define W_   64
#define HW_  4096
#define OC_  256
#define NPT  9          // K*K kernel points
#define KTOT 2304       // C_*NPT reduction length
#define HP1  65.0f      // Hp-1 = 66-1

#define POSB 64         // positions per block (4 N-tiles of 16)
#define SROW2 132       // paired sTile row stride in floats (2*64 + 4 pad)

__device__ __forceinline__ float sigmoidf_(float v) {
    return 1.0f / (1.0f + expf(-v));
}

// ---------------------------------------------------------------------------
// Kernel 1: compute the 30 control channels (18 offset, 3 ad, 9 mask) with a
// direct 3x3 conv, then emit final sampling coords px,py (clipped, masked)
// and modulated mask m into the workspace.  4 threads per position split the
// 256-channel reduction; LDS reduce.
// ---------------------------------------------------------------------------
__global__ __launch_bounds__(256) void dcn_prep(
    const float* __restrict__ x,
    const float* __restrict__ wp,   // (18,256,9)
    const float* __restrict__ bp,   // (18)
    const float* __restrict__ wm,   // (9,256,9)
    const float* __restrict__ wad,  // (3,256,9)
    float* __restrict__ pxw, float* __restrict__ pyw, float* __restrict__ mw)
{
    __shared__ float red[30 * 256];

    const int tid  = threadIdx.x;
    const int b    = blockIdx.y;
    const int pl   = tid & 63;       // position within 64-wide tile
    const int part = tid >> 6;       // 0..3 channel-split (wave-uniform)
    const int pg   = blockIdx.x * 64 + pl;
    const int i    = pg >> 6;
    const int j    = pg & 63;

    float acc[30];
#pragma unroll
    for (int o = 0; o < 30; ++o) acc[o] = 0.0f;

    const int cBeg = part * 64;
    for (int c = cBeg; c < cBeg + 64; ++c) {
        const float* xb = x + (((size_t)(b * C_ + c)) << 12);
        float xv[9];
#pragma unroll
        for (int t = 0; t < 9; ++t) {
            const int di = t / 3, dj = t % 3;
            const int ix = i + di - 1, iy = j + dj - 1;
            const bool ok = ((unsigned)ix < 64u) && ((unsigned)iy < 64u);
            xv[t] = ok ? xb[ix * 64 + iy] : 0.0f;
        }
        const size_t c9 = (size_t)c * 9;
#pragma unroll
        for (int o = 0; o < 18; ++o) {
            const float* w = wp + (size_t)o * KTOT + c9;
#pragma unroll
            for (int t = 0; t < 9; ++t) acc[o] = fmaf(w[t], xv[t], acc[o]);
        }
#pragma unroll
        for (int o = 0; o < 3; ++o) {
            const float* w = wad + (size_t)o * KTOT + c9;
#pragma unroll
            for (int t = 0; t < 9; ++t) acc[18 + o] = fmaf(w[t], xv[t], acc[18 + o]);
        }
#pragma unroll
        for (int o = 0; o < 9; ++o) {
            const float* w = wm + (size_t)o * KTOT + c9;
#pragma unroll
            for (int t = 0; t < 9; ++t) acc[21 + o] = fmaf(w[t], xv[t], acc[21 + o]);
        }
    }

#pragma unroll
    for (int o = 0; o < 30; ++o) red[o * 256 + tid] = acc[o];
    __syncthreads();

    if (tid < 64) {
        float fin[30];
#pragma unroll
        for (int o = 0; o < 30; ++o) {
            const float* r = red + o * 256 + tid;
            fin[o] = r[0] + r[64] + r[128] + r[192];
        }
#pragma unroll
        for (int o = 0; o < 18; ++o) fin[o] += bp[o];

#pragma unroll
        for (int n = 0; n < 9; ++n) {
            const float gx  = (float)(n / 3 - 1);
            const float gy  = (float)(n % 3 - 1);
            const float adb = 1.0f - sigmoidf_(fin[18 + (n % 3)]);
            float px0 = (float)(i + 1) + gx + fin[n]     + 2.0f * adb * gx;
            float py0 = (float)(j + 1) + gy + fin[9 + n] + 2.0f * adb * gy;
            const bool mx = (px0 < 1.0f) || (px0 > 64.0f);
            const bool my = (py0 < 1.0f) || (py0 > 64.0f);
            float pxm = mx ? floorf(px0) : px0;
            float pym = my ? floorf(py0) : py0;
            const float px = fminf(fmaxf(pxm, 0.0f), HP1);
            const float py = fminf(fmaxf(pym, 0.0f), HP1);
            const float mv = sigmoidf_(fin[21 + n]) * ((adb - 0.5f) * 4.0f);
            const int idx = (b * NPT + n) * HW_ + pg;
            pxw[idx] = px;
            pyw[idx] = py;
            mw[idx]  = mv;
        }
    }
}

// ---------------------------------------------------------------------------
// Kernel 2: main contraction via V_WMMA_F32_16X16X4_F32.
//   out[oc, p] = sum_{c,n} w_conv[oc, c*9+n] * m[n,p] * bilin(x[c], px, py)
// Block = 512 threads (16 waves): 64 positions x 256 output channels.
// Each wave owns one 16-oc row band and 4 position tiles, so every A-load
// (w_conv, global b64) feeds 4 WMMAs.  sTile stores K-pairs adjacently so
// each B operand is one ds_load_b64 into an aligned VGPR pair (no shuffle
// moves).  Bilinear params are packed int4/float4 -> 2x ds_load_b128 per
// sample in the gather phase.
// ---------------------------------------------------------------------------
__global__ __launch_bounds__(512) void dcn_main(
    const float* __restrict__ x,
    const float* __restrict__ wconv,  // (256, 2304)
    const float* __restrict__ pxw, const float* __restrict__ pyw,
    const float* __restrict__ mw,
    float* __restrict__ out)
{
    __shared__ float  sTile[72 * SROW2];   // [kpair][pos][2], pair-interleaved
    __shared__ int4   cOff4[POSB * 9];     // per (pos,n): 4 corner offsets
    __shared__ float4 cW4[POSB * 9];       // per (pos,n): 4 corner weights*m

    const int tid     = threadIdx.x;
    const int b       = blockIdx.y;
    const int posBase = blockIdx.x * POSB;
    const int lane    = tid & 31;
    const int wave    = tid >> 5;          // 0..15
    const int ocBase  = wave * 16;
    const int nlane   = lane & 15;
    const int klaneA  = (lane >> 4) << 1;  // 0 for lanes 0-15, 2 for 16-31

    // --- bilinear parameters for the 64 positions x 9 points -------------
    for (int e = tid; e < POSB * 9; e += 512) {
        const int n  = e % 9;
        const int pl = e / 9;
        const int pg = posBase + pl;
        const int idx = (b * NPT + n) * HW_ + pg;
        const float px = pxw[idx];
        const float py = pyw[idx];
        const float mv = mw[idx];
        const float flx = floorf(px), fly = floorf(py);
        const float qltx = fminf(fmaxf(flx,        0.0f), HP1);
        const float qlty = fminf(fmaxf(fly,        0.0f), HP1);
        const float qrbx = fminf(fmaxf(flx + 1.0f, 0.0f), HP1);
        const float qrby = fminf(fmaxf(fly + 1.0f, 0.0f), HP1);
        const float wlx = 1.0f + (qltx - px), wrx = 1.0f - (qrbx - px);
        const float wly = 1.0f + (qlty - py), wry = 1.0f - (qrby - py);
        const int ltx = (int)qltx, lty = (int)qlty;
        const int rbx = (int)qrbx, rby = (int)qrby;
        const int cx[4] = {ltx, rbx, ltx, rbx};
        const int cy[4] = {lty, rby, rby, lty};
        const float gw[4] = {wlx * wly, wrx * wry, wlx * wry, wrx * wly};
        int of[4];
        float wv[4];
#pragma unroll
        for (int c4 = 0; c4 < 4; ++c4) {
            const bool ok = (cx[c4] >= 1) && (cx[c4] <= 64) &&
                            (cy[c4] >= 1) && (cy[c4] <= 64);
            of[c4] = ok ? ((cx[c4] - 1) * 64 + (cy[c4] - 1)) : 0;
            wv[c4] = ok ? gw[c4] * mv : 0.0f;
        }
        cOff4[pl * 9 + n] = make_int4(of[0], of[1], of[2], of[3]);
        cW4[pl * 9 + n]   = make_float4(wv[0], wv[1], wv[2], wv[3]);
    }
    __syncthreads();

    v8f acc[4];
#pragma unroll
    for (int pt = 0; pt < 4; ++pt) acc[pt] = {};

    for (int c0 = 0; c0 < C_; c0 += 16) {
        // ---- build modulated sample tile (K-pair interleaved) ------------
        for (int e = tid; e < 144 * POSB; e += 512) {
            const int kl = e >> 6;          // 0..143
            const int pl = e & 63;          // 0..63 (constant per thread)
            const int ch = kl / 9;
            const int n  = kl - ch * 9;
            const float* xb = x + (((size_t)(b * C_ + c0 + ch)) << 12);
            const int4   o4 = cOff4[pl * 9 + n];   // ds_load_b128
            const float4 w4 = cW4[pl * 9 + n];     // ds_load_b128
            const float s = w4.x * xb[o4.x] + w4.y * xb[o4.y]
                          + w4.z * xb[o4.z] + w4.w * xb[o4.w];
            sTile[(kl >> 1) * SROW2 + pl * 2 + (kl & 1)] = s;
        }
        __syncthreads();

        // ---- 144 WMMAs over this 144-deep chunk (4 N-tiles per A-load) ---
        const float* wbase = wconv + (size_t)(ocBase + nlane) * KTOT
                                   + c0 * 9 + klaneA;
#pragma unroll 2
        for (int kk = 0; kk < 144; kk += 4) {
            const v2f a = *(const v2f*)(wbase + kk);    // A: 16x4 f32 layout
            const int kp = (kk + klaneA) >> 1;          // even pair index
            const float* srow = &sTile[kp * SROW2 + nlane * 2];
#pragma unroll
            for (int pt = 0; pt < 4; ++pt) {
                const v2f bb = *(const v2f*)(srow + pt * 32);  // ds_load_b64
                acc[pt] = __builtin_amdgcn_wmma_f32_16x16x4_f32(
                    false, a, false, bb, (short)0, acc[pt], false, false);
            }
        }
        __syncthreads();
    }

    // ---- store D (16x16 f32): VGPR r -> M = r (+8 for upper half-wave) ---
    const int mhalf = (lane >> 4) << 3;
#pragma unroll
    for (int pt = 0; pt < 4; ++pt) {
#pragma unroll
        for (int r = 0; r < 8; ++r) {
            const int oc = ocBase + mhalf + r;
            out[(size_t)(b * OC_ + oc) * HW_ + posBase + pt * 16 + nlane] =
                acc[pt][r];
        }
    }
}

extern "C" void kernel_launch(void* const* d_in, const int* in_sizes, int n_in,
                              void* d_out, int out_size, void* d_ws, size_t ws_size,
                              hipStream_t stream) {
    const float* x      = (const float*)d_in[0];
    const float* w_p    = (const float*)d_in[1];
    const float* b_p    = (const float*)d_in[2];
    const float* w_m    = (const float*)d_in[3];
    const float* w_ad   = (const float*)d_in[4];
    const float* w_conv = (const float*)d_in[5];
    float* out = (float*)d_out;

    float* ws  = (float*)d_ws;
    const size_t planeN = (size_t)B_ * NPT * HW_;   // 147456 floats
    float* pxw = ws;
    float* pyw = ws + planeN;
    float* mw  = ws + 2 * planeN;

    // Phase 1: control channels -> sampling coords + modulated mask
    dcn_prep<<<dim3(64, B_), 256, 0, stream>>>(x, w_p, b_p, w_m, w_ad, pxw, pyw, mw);

    // Phase 2: WMMA contraction (64 pos x 256 oc per block)
    dcn_main<<<dim3(HW_ / POSB, B_), 512, 0, stream>>>(x, w_conv, pxw, pyw, mw, out);
}